// GraphAutoEncoderModel_40329742909510
// MI455X (gfx1250) — compile-verified
//
#include <hip/hip_runtime.h>
#include <stdint.h>

// Problem constants (from the reference)
#define S1N          8
#define S2N          8
#define FDIM         128     // feature columns
#define ROWS_PER_ID  288     // 2 branches * 8 * (1+8+1+8)
#define ROW_W        130     // [w | 128 feats | wc]
#define WAVES        8       // 256 threads / wave32
#define BUFS         9       // rows per half-group: 1 hub-1 + 8 hub-2

// ---- CDNA5 async global<->LDS data movement (ASYNCcnt-tracked) ----------
// Low 32 bits of a flat pointer into __shared__ == workgroup-relative LDS
// byte offset (flat LDS aperture keeps the offset in addr[31:0]).
__device__ __forceinline__ uint32_t lds_off(const void* p) {
    return (uint32_t)(uintptr_t)p;
}

__device__ __forceinline__ void async_load_b128(uint32_t lds, const float* g) {
    // copies 16 bytes from global address g to LDS offset lds; RT policy (keep features hot in L2)
    asm volatile("global_load_async_to_lds_b128 %0, %1, off"
                 :: "v"(lds), "v"(g) : "memory");
}

// IOFFSET is added to BOTH the LDS and global addresses (ISA 08_async §4.4),
// and both sides advance by 128B per burst -> fold the burst index into the
// immediate and reuse one base address pair per output row.
template <int OFF>
__device__ __forceinline__ void async_store_b32_nt(float* g, uint32_t lds) {
    asm volatile("global_store_async_from_lds_b32 %0, %1, off offset:%2 th:TH_STORE_NT"
                 :: "v"(g), "v"(lds), "i"(OFF) : "memory");
}

__device__ __forceinline__ void wait_async0() {
    asm volatile("s_wait_asynccnt 0" ::: "memory");
}

// -------------------------------------------------------------------------
// Row layout per seed b: j in [0,288).
//   branch = j>=144 ('out' hub-1), jj = j%144, s1 = jj/18, t = jj%18
//   t==0 / t==9         : [w1   | features[n1]               | w1*w1  ]
//   t in 1..8  (s2=t-1) : [w2in | features[in_samp[n1,s2]]   | w2in*w1]
//   t in 10..17(s2=t-10): [w2o  | features[out_samp[n1,s2]]  | w2o*w1 ]
// Wave w owns branch (w>=4) and s1 in {2*(w%4), 2*(w%4)+1}: all row decode
// is compile-time constant inside the unrolled loops below.
__global__ __launch_bounds__(256)
void gae_gather_kernel(const float* __restrict__ features,     // (N,128)
                       const float* __restrict__ in_amnt,      // (N,8,1)
                       const float* __restrict__ out_amnt,     // (N,8,1)
                       const int*   __restrict__ in_samp,      // (N,8)
                       const int*   __restrict__ out_samp,     // (N,8)
                       const int*   __restrict__ node_ids,     // (B,)
                       float*       __restrict__ out)          // (B,288,130)
{
    __shared__ float smem[WAVES][BUFS][FDIM];   // 36 KB / block

    const int tid  = (int)threadIdx.x;
    const int wave = tid >> 5;                  // wave32 (gfx1250)
    const int lane = tid & 31;
    const int b    = (int)blockIdx.x;

    const int node = node_ids[b];

    const bool br_out  = (wave >= 4);
    const int  s1base  = (wave & 3) * 2;
    const int  jbranch = br_out ? 144 : 0;
    const int*   samp1 = br_out ? out_samp : in_samp;
    const float* am1   = br_out ? out_amnt : in_amnt;

    // Loop-invariant LDS byte offsets: load side (b128, lane*16) and store
    // side (b32 bursts, lane*4). Hoisted once, live in 18 VGPRs.
    uint32_t lds_ld[BUFS], lds_st[BUFS];
    #pragma unroll
    for (int r = 0; r < BUFS; ++r) {
        lds_ld[r] = lds_off(&smem[wave][r][lane * 4]);
        lds_st[r] = lds_off(&smem[wave][r][lane]);
    }

    #pragma unroll
    for (int g = 0; g < 2; ++g) {               // two s1 groups per wave
        const int   s1 = s1base + g;
        const int   n1 = samp1[node * S1N + s1];
        const float w1 = am1  [node * S1N + s1];

        #pragma unroll
        for (int half = 0; half < 2; ++half) {  // hub-2 'in' then 'out'
            const int*   samp2 = half ? out_samp : in_samp;
            const float* am2   = half ? out_amnt : in_amnt;

            // 8 contiguous hub-2 neighbor ids / weights: 2+2 b128 loads
            const int4*   np = (const int4*)  (samp2 + (size_t)n1 * S2N);
            const float4* wp = (const float4*)(am2   + (size_t)n1 * S2N);
            const int4   na = np[0], nb2 = np[1];
            const float4 wa = wp[0], wb  = wp[1];

            int   src[BUFS];
            float wl[BUFS];
            src[0] = n1;    wl[0] = w1;
            src[1] = na.x;  wl[1] = wa.x;
            src[2] = na.y;  wl[2] = wa.y;
            src[3] = na.z;  wl[3] = wa.z;
            src[4] = na.w;  wl[4] = wa.w;
            src[5] = nb2.x; wl[5] = wb.x;
            src[6] = nb2.y; wl[6] = wb.y;
            src[7] = nb2.z; wl[7] = wb.z;
            src[8] = nb2.w; wl[8] = wb.w;

            // ---- enqueue 9 async row gathers (512B each, 32x b128) ----
            #pragma unroll
            for (int r = 0; r < BUFS; ++r) {
                async_load_b128(lds_ld[r],
                                features + (size_t)src[r] * FDIM + lane * 4);
            }

            wait_async0();   // rows resident in LDS

            // ---- coalesced NT async stores out of LDS + scalars ----
            const int j0 = jbranch + s1 * 18 + half * 9;   // t = half*9 + r
            #pragma unroll
            for (int r = 0; r < BUFS; ++r) {
                float* grow = out + ((size_t)b * ROWS_PER_ID + (j0 + r))
                                      * ROW_W;
                if (lane == 0) {
                    const float wc = (r == 0) ? (w1 * w1) : (wl[r] * w1);
                    __builtin_nontemporal_store(wl[r], grow);       // k=0
                    __builtin_nontemporal_store(wc,    grow + 129); // k=129
                }
                // 128 feats at +1 (4B-aligned): one base address, four
                // fully-coalesced b32 bursts via the shared IOFFSET field.
                float* gb = grow + 1 + lane;
                async_store_b32_nt<0>  (gb, lds_st[r]);
                async_store_b32_nt<128>(gb, lds_st[r]);
                async_store_b32_nt<256>(gb, lds_st[r]);
                async_store_b32_nt<384>(gb, lds_st[r]);
            }

            wait_async0();   // stores consumed LDS before buffer reuse
        }
    }
}

// -------------------------------------------------------------------------
extern "C" void kernel_launch(void* const* d_in, const int* in_sizes, int n_in,
                              void* d_out, int out_size, void* d_ws, size_t ws_size,
                              hipStream_t stream) {
    const float* features = (const float*)d_in[0];
    const float* in_amnt  = (const float*)d_in[1];
    const float* out_amnt = (const float*)d_in[2];
    const int*   in_samp  = (const int*)  d_in[3];
    const int*   out_samp = (const int*)  d_in[4];
    const int*   node_ids = (const int*)  d_in[5];

    const int B = in_sizes[5];   // 2048 seed nodes -> one block each

    gae_gather_kernel<<<dim3(B), dim3(256), 0, stream>>>(
        features, in_amnt, out_amnt, in_samp, out_samp, node_ids,
        (float*)d_out);
}